// SfeduModel_1511828489036
// MI455X (gfx1250) — compile-verified
//
#include <hip/hip_runtime.h>
#include <math.h>

#define N_NODES  200000
#define N_EDGES  6400000
#define N_GRAPHS 256
#define D_IN     128
#define CH       32
#define N_LABELS 10

typedef __attribute__((ext_vector_type(2))) float v2f;
typedef __attribute__((ext_vector_type(8))) float v8f;

__device__ __forceinline__ void atomic_add_f32(float* p, float v) {
    unsafeAtomicAdd(p, v);   // hardware global_atomic_add_f32
}

// ---------------------------------------------------------------------------
// GEMM: D[M x 32] = A[M x K] * B[K x 32] using V_WMMA_F32_16X16X4_F32.
// One wave computes a 16-row x 32-col tile (two 16x16 accumulators).
// ISA layouts (wave32): lane = 16*h + L, h in {0,1}, L in [0,16)
//   A frag v2f: a[v] = A[L][k + v + 2h]
//   B frag v2f: b[v] = B[k + v + 2h][n]          (n = L or L+16)
//   C/D  v8f : c[r] = D[r + 8h][n]
// ---------------------------------------------------------------------------
template <int K>
__global__ __launch_bounds__(256)
void gemm_wmma_f32(const float* __restrict__ A, const float* __restrict__ B,
                   float* __restrict__ D, int mtiles) {
    const int wave = (int)((blockIdx.x * blockDim.x + threadIdx.x) >> 5);
    if (wave >= mtiles) return;                    // uniform per wave: EXEC all-1s
    const int lane = (int)(threadIdx.x & 31);
    const int L = lane & 15;
    const int h = lane >> 4;

    const float* arow = A + (size_t)(wave * 16 + L) * K;
    v8f c0 = {};
    v8f c1 = {};

#pragma unroll 4
    for (int k = 0; k < K; k += 4) {
        v2f a;
        a.x = arow[k + 2 * h];
        a.y = arow[k + 2 * h + 1];

        const float* bp = B + (size_t)(k + 2 * h) * CH + L;
        v2f b0, b1;
        b0.x = bp[0];
        b0.y = bp[CH];
        b1.x = bp[16];
        b1.y = bp[CH + 16];

        c0 = __builtin_amdgcn_wmma_f32_16x16x4_f32(false, a, false, b0,
                                                   (short)0, c0, false, false);
        c1 = __builtin_amdgcn_wmma_f32_16x16x4_f32(false, a, false, b1,
                                                   (short)0, c1, false, false);
    }

    float* dp = D + (size_t)(wave * 16 + 8 * h) * CH + L;
#pragma unroll
    for (int r = 0; r < 8; ++r) {
        dp[(size_t)r * CH]      = c0[r];
        dp[(size_t)r * CH + 16] = c1[r];
    }
}

// ---------------------------------------------------------------------------
// Zero-fill (float4 stores)
// ---------------------------------------------------------------------------
__global__ __launch_bounds__(256)
void zero_f4(float4* __restrict__ p, long long n4) {
    long long i = (long long)blockIdx.x * blockDim.x + threadIdx.x;
    if (i < n4) p[i] = make_float4(0.f, 0.f, 0.f, 0.f);
}

// ---------------------------------------------------------------------------
// Edge scatter: one wave per edge; lane c handles channel c.
//   agg[dst, c] += h[src, c] * w[e]
// 128B coalesced gather + 128B contiguous fp32 atomics per edge.
// ---------------------------------------------------------------------------
__global__ __launch_bounds__(256)
void edge_scatter(const float* __restrict__ h, const int* __restrict__ src,
                  const int* __restrict__ dst, const float* __restrict__ ew,
                  float* __restrict__ agg, int n_edges) {
    long long t = (long long)blockIdx.x * blockDim.x + threadIdx.x;
    int c = (int)(t & 31);
    long long e = t >> 5;
    if (e >= n_edges) return;
    int s = src[e];
    int d = dst[e];
    float w = ew[e];
    float v = h[(size_t)s * CH + c] * w;
    atomic_add_f32(&agg[(size_t)d * CH + c], v);
}

// ---------------------------------------------------------------------------
// In-place bias + relu over [N_NODES, CH]
// ---------------------------------------------------------------------------
__global__ __launch_bounds__(256)
void bias_relu(float* __restrict__ a, const float* __restrict__ b, long long n) {
    long long i = (long long)blockIdx.x * blockDim.x + threadIdx.x;
    if (i >= n) return;
    float v = a[i] + b[i & (CH - 1)];
    a[i] = v > 0.f ? v : 0.f;
}

// ---------------------------------------------------------------------------
// Segment pooling: wave per node; lanes = channels. counts via lane 0.
// ---------------------------------------------------------------------------
__global__ __launch_bounds__(256)
void pool_scatter(const float* __restrict__ x, const int* __restrict__ seg,
                  float* __restrict__ sums, float* __restrict__ counts, int n_nodes) {
    long long t = (long long)blockIdx.x * blockDim.x + threadIdx.x;
    int c = (int)(t & 31);
    long long i = t >> 5;
    if (i >= n_nodes) return;
    int g = seg[i];
    atomic_add_f32(&sums[(size_t)g * CH + c], x[(size_t)i * CH + c]);
    if (c == 0) atomic_add_f32(&counts[g], 1.0f);
}

// ---------------------------------------------------------------------------
// Head: pooled = sums/max(counts,1); logits = pooled @ Wd + bd; softmax.
// One thread per graph (256 graphs -> one block).
// ---------------------------------------------------------------------------
__global__ __launch_bounds__(256)
void head_softmax(const float* __restrict__ sums, const float* __restrict__ counts,
                  const float* __restrict__ Wd, const float* __restrict__ bd,
                  float* __restrict__ out) {
    int g = (int)threadIdx.x;
    if (g >= N_GRAPHS) return;
    float inv = 1.0f / fmaxf(counts[g], 1.0f);
    float p[CH];
#pragma unroll
    for (int c = 0; c < CH; ++c) p[c] = sums[(size_t)g * CH + c] * inv;

    float logit[N_LABELS];
    float mx = -INFINITY;
#pragma unroll
    for (int j = 0; j < N_LABELS; ++j) {
        float acc = bd[j];
#pragma unroll
        for (int c = 0; c < CH; ++c) acc += p[c] * Wd[c * N_LABELS + j];
        logit[j] = acc;
        mx = fmaxf(mx, acc);
    }
    float s = 0.f;
#pragma unroll
    for (int j = 0; j < N_LABELS; ++j) {
        logit[j] = __expf(logit[j] - mx);
        s += logit[j];
    }
    float rs = 1.0f / s;
#pragma unroll
    for (int j = 0; j < N_LABELS; ++j) out[(size_t)g * N_LABELS + j] = logit[j] * rs;
}

// ---------------------------------------------------------------------------
extern "C" void kernel_launch(void* const* d_in, const int* in_sizes, int n_in,
                              void* d_out, int out_size, void* d_ws, size_t ws_size,
                              hipStream_t stream) {
    const float* x    = (const float*)d_in[0];
    const int*   esrc = (const int*)  d_in[1];
    const int*   edst = (const int*)  d_in[2];
    const float* ew   = (const float*)d_in[3];
    const int*   seg  = (const int*)  d_in[4];
    const float* W1   = (const float*)d_in[5];
    const float* b1   = (const float*)d_in[6];
    const float* W2   = (const float*)d_in[7];
    const float* b2   = (const float*)d_in[8];
    const float* W3   = (const float*)d_in[9];
    const float* b3   = (const float*)d_in[10];
    const float* Wd   = (const float*)d_in[11];
    const float* bd   = (const float*)d_in[12];
    float* out = (float*)d_out;

    const long long nfeat = (long long)N_NODES * CH;     // 6.4M floats
    float* bufA   = (float*)d_ws;                        // GEMM output / message src
    float* bufB   = bufA + nfeat;                        // scatter target / next input
    float* sums   = bufB + nfeat;                        // [N_GRAPHS, CH]
    float* counts = sums + (size_t)N_GRAPHS * CH;        // [N_GRAPHS]

    const int mtiles      = N_NODES / 16;                            // 12500
    const int gemm_blocks = (mtiles + 7) / 8;                        // 8 waves/block
    const long long et    = (long long)N_EDGES * 32;
    const int edge_blocks = (int)((et + 255) / 256);
    const int ew_blocks   = (int)((nfeat + 255) / 256);
    const int z_blocks    = (int)((nfeat / 4 + 255) / 256);
    const long long pt    = (long long)N_NODES * 32;
    const int pool_blocks = (int)((pt + 255) / 256);

    // ---- layer 1: h = x @ W1 ; agg = scatter(h) ; x1 = relu(agg + b1) ----
    gemm_wmma_f32<D_IN><<<gemm_blocks, 256, 0, stream>>>(x, W1, bufA, mtiles);
    zero_f4<<<z_blocks, 256, 0, stream>>>((float4*)bufB, nfeat / 4);
    edge_scatter<<<edge_blocks, 256, 0, stream>>>(bufA, esrc, edst, ew, bufB, N_EDGES);
    bias_relu<<<ew_blocks, 256, 0, stream>>>(bufB, b1, nfeat);

    // ---- layer 2 ----
    gemm_wmma_f32<CH><<<gemm_blocks, 256, 0, stream>>>(bufB, W2, bufA, mtiles);
    zero_f4<<<z_blocks, 256, 0, stream>>>((float4*)bufB, nfeat / 4);
    edge_scatter<<<edge_blocks, 256, 0, stream>>>(bufA, esrc, edst, ew, bufB, N_EDGES);
    bias_relu<<<ew_blocks, 256, 0, stream>>>(bufB, b2, nfeat);

    // ---- layer 3 ----
    gemm_wmma_f32<CH><<<gemm_blocks, 256, 0, stream>>>(bufB, W3, bufA, mtiles);
    zero_f4<<<z_blocks, 256, 0, stream>>>((float4*)bufB, nfeat / 4);
    edge_scatter<<<edge_blocks, 256, 0, stream>>>(bufA, esrc, edst, ew, bufB, N_EDGES);
    bias_relu<<<ew_blocks, 256, 0, stream>>>(bufB, b3, nfeat);

    // ---- pool + head ----
    const long long psz = (long long)N_GRAPHS * CH + N_GRAPHS;       // 8448 floats
    zero_f4<<<(int)((psz / 4 + 255) / 256), 256, 0, stream>>>((float4*)sums, psz / 4);
    pool_scatter<<<pool_blocks, 256, 0, stream>>>(bufB, seg, sums, counts, N_NODES);
    head_softmax<<<1, 256, 0, stream>>>(sums, counts, Wd, bd, out);
}